// LSTMHetGNN_49752901157177
// MI455X (gfx1250) — compile-verified
//
#include <hip/hip_runtime.h>

// Problem constants (match reference)
#define NN1 20000
#define NN2 20000
#define NEDGE 320000
#define IN_DIM 64
#define HID 128
#define OUTD 128
#define GATES 512  // 4*HID

typedef float v2f __attribute__((ext_vector_type(2)));
typedef float v8f __attribute__((ext_vector_type(8)));

__device__ __forceinline__ v8f wmma4(v2f a, v2f b, v8f c) {
  // exact fp32 MAC on the CDNA5 matrix pipe: V_WMMA_F32_16X16X4_F32
  return __builtin_amdgcn_wmma_f32_16x16x4_f32(false, a, false, b, (short)0, c,
                                               false, false);
}

// ---------------------------------------------------------------------------
// out[M,512] = A[M,64] @ W[512,64]^T + (b1+b2)   (LSTM input projection)
// One wave owns a 16-row x 128-col strip: 8 accumulator tiles, so each A
// fragment is loaded once and reused by 8 WMMAs (8x less A traffic).
// ---------------------------------------------------------------------------
__global__ void xw_gemm_kernel(const float* __restrict__ A,
                               const float* __restrict__ W,
                               const float* __restrict__ b1,
                               const float* __restrict__ b2,
                               float* __restrict__ out, int M) {
  const int NCG = GATES / 128;  // 4 column groups of 128
  int wave = (blockIdx.x * blockDim.x + threadIdx.x) >> 5;
  int lane = threadIdx.x & 31;
  int tm = wave / NCG, tg = wave % NCG;
  if (tm * 16 >= M) return;  // wave-uniform, EXEC stays all-ones
  int rowBase = tm * 16, colBase = tg * 128;
  int l15 = lane & 15;
  int kof = (lane >> 4) << 1;  // 0 or 2
  const float* Arow = A + (size_t)(rowBase + l15) * IN_DIM + kof;
  const float* Wrow = W + (size_t)(colBase + l15) * IN_DIM + kof;
  v8f c[8];
#pragma unroll
  for (int t = 0; t < 8; ++t) c[t] = {};
#pragma unroll
  for (int kk = 0; kk < IN_DIM; kk += 4) {
    v2f a;
    a.x = Arow[kk];
    a.y = Arow[kk + 1];
#pragma unroll
    for (int t = 0; t < 8; ++t) {
      v2f b;
      b.x = Wrow[(size_t)t * 16 * IN_DIM + kk];
      b.y = Wrow[(size_t)t * 16 * IN_DIM + kk + 1];
      c[t] = wmma4(a, b, c[t]);
    }
  }
  int mb = rowBase + ((lane >> 4) << 3);
#pragma unroll
  for (int t = 0; t < 8; ++t) {
    int n = colBase + t * 16 + l15;
    float bias = b1[n] + b2[n];
#pragma unroll
    for (int i = 0; i < 8; ++i)
      out[(size_t)(mb + i) * GATES + n] = c[t][i] + bias;
  }
}

// ---------------------------------------------------------------------------
// Fused SAGE linear: out = A1@W1^T + A2@W2^T + bias (optional ReLU)
// A1/A2: [M,128], W1/W2: [128,128]. One wave per 16-row x 128-col strip;
// A fragment loaded once per k-step, reused across 8 column tiles.
// ---------------------------------------------------------------------------
__global__ void sage_gemm_kernel(const float* __restrict__ A1,
                                 const float* __restrict__ W1,
                                 const float* __restrict__ A2,
                                 const float* __restrict__ W2,
                                 const float* __restrict__ bias,
                                 float* __restrict__ out, int M, int relu) {
  int wave = (blockIdx.x * blockDim.x + threadIdx.x) >> 5;
  int lane = threadIdx.x & 31;
  int tm = wave;  // one 16-row strip per wave, full 128-col width
  if (tm * 16 >= M) return;
  int rowBase = tm * 16;
  int l15 = lane & 15;
  int kof = (lane >> 4) << 1;
  v8f c[8];
#pragma unroll
  for (int t = 0; t < 8; ++t) c[t] = {};
  {
    const float* a1 = A1 + (size_t)(rowBase + l15) * HID + kof;
    const float* w1 = W1 + (size_t)l15 * HID + kof;
#pragma unroll
    for (int kk = 0; kk < HID; kk += 4) {
      v2f a;
      a.x = a1[kk];
      a.y = a1[kk + 1];
#pragma unroll
      for (int t = 0; t < 8; ++t) {
        v2f b;
        b.x = w1[(size_t)t * 16 * HID + kk];
        b.y = w1[(size_t)t * 16 * HID + kk + 1];
        c[t] = wmma4(a, b, c[t]);
      }
    }
  }
  {
    const float* a2 = A2 + (size_t)(rowBase + l15) * HID + kof;
    const float* w2 = W2 + (size_t)l15 * HID + kof;
#pragma unroll
    for (int kk = 0; kk < HID; kk += 4) {
      v2f a;
      a.x = a2[kk];
      a.y = a2[kk + 1];
#pragma unroll
      for (int t = 0; t < 8; ++t) {
        v2f b;
        b.x = w2[(size_t)t * 16 * HID + kk];
        b.y = w2[(size_t)t * 16 * HID + kk + 1];
        c[t] = wmma4(a, b, c[t]);
      }
    }
  }
  int mb = rowBase + ((lane >> 4) << 3);
#pragma unroll
  for (int t = 0; t < 8; ++t) {
    int n = t * 16 + l15;
    float bv = bias[n];
#pragma unroll
    for (int i = 0; i < 8; ++i) {
      float v = c[t][i] + bv;
      if (relu) v = fmaxf(v, 0.0f);
      out[(size_t)(mb + i) * OUTD + n] = v;
    }
  }
}

// ---------------------------------------------------------------------------
// Sequential LSTM scan. One 512-thread block per sequence (16 wave32s).
// Each thread keeps its own Whh row (128 floats) in VGPRs; only h[128]
// lives in LDS and is read as same-address broadcasts (ds_load_b128-able).
// Thread j computes gate pre-activation j; threads 0..127 update c/h.
// ---------------------------------------------------------------------------
__device__ __forceinline__ float fast_sigmoid(float x) {
  return 1.0f / (1.0f + __expf(-x));
}
__device__ __forceinline__ float fast_tanh(float x) {
  float e = __expf(-2.0f * x);
  return (1.0f - e) / (1.0f + e);
}

__global__ void lstm_scan_kernel(const float* __restrict__ xw1,
                                 const float* __restrict__ xw2,
                                 const float* __restrict__ Whh,
                                 float* __restrict__ h1out,
                                 float* __restrict__ h2out, int T) {
  __shared__ float hb[HID];
  __shared__ float cb[HID];
  __shared__ float gb[GATES];
  const float* xw = (blockIdx.x == 0) ? xw1 : xw2;
  float* hout = (blockIdx.x == 0) ? h1out : h2out;
  int j = threadIdx.x;
  // register-resident weight row: Whh[j][0..127]
  float w[HID];
#pragma unroll
  for (int k = 0; k < HID; ++k) w[k] = Whh[(size_t)j * HID + k];
  if (j < HID) {
    hb[j] = 0.0f;
    cb[j] = 0.0f;
  }
  __syncthreads();
  int gate = j >> 7;  // 0:i 1:f 2:g 3:o (torch gate order)
  for (int t = 0; t < T; ++t) {
    float acc = xw[(size_t)t * GATES + j];
#pragma unroll
    for (int k = 0; k < HID; ++k) acc = fmaf(hb[k], w[k], acc);
    gb[j] = (gate == 2) ? fast_tanh(acc) : fast_sigmoid(acc);
    __syncthreads();
    if (j < HID) {
      float cn = gb[HID + j] * cb[j] + gb[j] * gb[2 * HID + j];
      cb[j] = cn;
      float hn = gb[3 * HID + j] * fast_tanh(cn);
      hb[j] = hn;
      hout[(size_t)t * HID + j] = hn;
    }
    __syncthreads();
  }
}

// ---------------------------------------------------------------------------
// Edge scatter: sum[dst] += feat[src]; cnt[dst] += 1.  2 edges per 256-thr blk.
// ---------------------------------------------------------------------------
__global__ void scatter_kernel(const float* __restrict__ feat,
                               const int* __restrict__ edges,  // [2,E] flat
                               float* __restrict__ sum,
                               float* __restrict__ cnt) {
  int e = blockIdx.x * 2 + (threadIdx.x >> 7);
  int f = threadIdx.x & 127;
  int s = edges[e];
  int d = edges[NEDGE + e];
  atomicAdd(&sum[(size_t)d * HID + f], feat[(size_t)s * HID + f]);
  if (f == 0) atomicAdd(&cnt[d], 1.0f);
}

// sum /= max(cnt,1)  (in place -> mean)
__global__ void mean_kernel(float* __restrict__ sum,
                            const float* __restrict__ cnt, int n) {
  int i = blockIdx.x * 2 + (threadIdx.x >> 7);
  int f = threadIdx.x & 127;
  if (i < n) sum[(size_t)i * HID + f] /= fmaxf(cnt[i], 1.0f);
}

// ---------------------------------------------------------------------------
extern "C" void kernel_launch(void* const* d_in, const int* in_sizes, int n_in,
                              void* d_out, int out_size, void* d_ws,
                              size_t ws_size, hipStream_t stream) {
  const float* x1 = (const float*)d_in[0];
  const float* x2 = (const float*)d_in[1];
  const int* e1 = (const int*)d_in[2];  // type1 -> type2
  const int* e2 = (const int*)d_in[3];  // type2 -> type1
  const float* Wih = (const float*)d_in[4];
  const float* Whh = (const float*)d_in[5];
  const float* bih = (const float*)d_in[6];
  const float* bhh = (const float*)d_in[7];
  const float* c1e1_Wl = (const float*)d_in[8];
  const float* c1e1_bl = (const float*)d_in[9];
  const float* c1e1_Wr = (const float*)d_in[10];
  const float* c1e2_Wl = (const float*)d_in[11];
  const float* c1e2_bl = (const float*)d_in[12];
  const float* c1e2_Wr = (const float*)d_in[13];
  const float* c2e1_Wl = (const float*)d_in[14];
  const float* c2e1_bl = (const float*)d_in[15];
  const float* c2e1_Wr = (const float*)d_in[16];
  const float* c2e2_Wl = (const float*)d_in[17];
  const float* c2e2_bl = (const float*)d_in[18];
  const float* c2e2_Wr = (const float*)d_in[19];

  float* out = (float*)d_out;
  float* f1 = out;                      // [N1,128]
  float* f2 = out + (size_t)NN1 * HID;  // [N2,128]

  // workspace layout (floats)
  float* ws = (float*)d_ws;
  float* xw1 = ws;
  float* xw2 = xw1 + (size_t)NN1 * GATES;
  float* h1 = xw2 + (size_t)NN2 * GATES;
  float* h2 = h1 + (size_t)NN1 * HID;
  float* agg1 = h2 + (size_t)NN2 * HID;  // segment mean, dst = type1
  float* agg2 = agg1 + (size_t)NN1 * HID;
  float* o1 = agg2 + (size_t)NN2 * HID;
  float* o2 = o1 + (size_t)NN1 * HID;
  float* cnt1 = o2 + (size_t)NN2 * HID;
  float* cnt2 = cnt1 + NN1;

  // ---- Phase 1: LSTM input projection (WMMA fp32 GEMM) ----
  {
    int waves = (NN1 / 16) * (GATES / 128);  // 1250 * 4 = 5000
    dim3 grd(waves / 8), blk(256);
    xw_gemm_kernel<<<grd, blk, 0, stream>>>(x1, Wih, bih, bhh, xw1, NN1);
    xw_gemm_kernel<<<grd, blk, 0, stream>>>(x2, Wih, bih, bhh, xw2, NN2);
  }

  // ---- Phase 2: sequential LSTM scan, Whh rows register-resident ----
  lstm_scan_kernel<<<2, GATES, 0, stream>>>(xw1, xw2, Whh, h1, h2, NN1);

  // ---- Phase 3: conv1 (segment mean + fused SAGE GEMM + ReLU) ----
  hipMemsetAsync(agg1, 0, (size_t)NN1 * HID * sizeof(float), stream);
  hipMemsetAsync(agg2, 0, (size_t)NN2 * HID * sizeof(float), stream);
  hipMemsetAsync(cnt1, 0, NN1 * sizeof(float), stream);
  hipMemsetAsync(cnt2, 0, NN2 * sizeof(float), stream);
  scatter_kernel<<<NEDGE / 2, 256, 0, stream>>>(h1, e1, agg2, cnt2);
  scatter_kernel<<<NEDGE / 2, 256, 0, stream>>>(h2, e2, agg1, cnt1);
  mean_kernel<<<NN2 / 2, 256, 0, stream>>>(agg2, cnt2, NN2);
  mean_kernel<<<NN1 / 2, 256, 0, stream>>>(agg1, cnt1, NN1);
  {
    int waves = NN2 / 16;  // 1250 row strips
    dim3 grd((waves + 7) / 8), blk(256);
    sage_gemm_kernel<<<grd, blk, 0, stream>>>(agg2, c1e1_Wl, h2, c1e1_Wr,
                                              c1e1_bl, o2, NN2, 1);
    sage_gemm_kernel<<<grd, blk, 0, stream>>>(agg1, c1e2_Wl, h1, c1e2_Wr,
                                              c1e2_bl, o1, NN1, 1);
  }

  // ---- Phase 4: conv2 ----
  hipMemsetAsync(agg1, 0, (size_t)NN1 * HID * sizeof(float), stream);
  hipMemsetAsync(agg2, 0, (size_t)NN2 * HID * sizeof(float), stream);
  hipMemsetAsync(cnt1, 0, NN1 * sizeof(float), stream);
  hipMemsetAsync(cnt2, 0, NN2 * sizeof(float), stream);
  scatter_kernel<<<NEDGE / 2, 256, 0, stream>>>(o1, e1, agg2, cnt2);
  scatter_kernel<<<NEDGE / 2, 256, 0, stream>>>(o2, e2, agg1, cnt1);
  mean_kernel<<<NN2 / 2, 256, 0, stream>>>(agg2, cnt2, NN2);
  mean_kernel<<<NN1 / 2, 256, 0, stream>>>(agg1, cnt1, NN1);
  {
    int waves = NN2 / 16;
    dim3 grd((waves + 7) / 8), blk(256);
    sage_gemm_kernel<<<grd, blk, 0, stream>>>(agg2, c2e1_Wl, o2, c2e1_Wr,
                                              c2e1_bl, f2, NN2, 0);
    sage_gemm_kernel<<<grd, blk, 0, stream>>>(agg1, c2e2_Wl, o1, c2e2_Wr,
                                              c2e2_bl, f1, NN1, 0);
  }
}